// CRF_65077344469312
// MI455X (gfx1250) — compile-verified
//
#include <hip/hip_runtime.h>
#include <hip/hip_bf16.h>
#include <math.h>

typedef float v2f __attribute__((ext_vector_type(2)));
typedef float v8f __attribute__((ext_vector_type(8)));

#define B_ 64
#define T_ 4096
#define D_ 256
#define N_ 64
#define WPAD_ (D_ + 4)            // stride 260 -> conflict-free b64 LDS reads
#define WT_BYTES_ (N_ * WPAD_ * 4)  // 66,560 B dynamic LDS

// ---------------------------------------------------------------------------
// Kernel 1: pot[b,t,n] = x[b,t,:] @ W[:,n] + b[n] (+ boundaries at t=0 / T-1)
// fp32 WMMA 16x16x4. One wave computes a 16(M) x 64(N) tile so x is read once.
// W is staged TRANSPOSED (+pad) in dynamic LDS so each B fragment is one
// aligned ds_load_b64 straight into an even VGPR pair (no repack moves).
// ---------------------------------------------------------------------------
__global__ __launch_bounds__(128) void proj_wmma_kernel(
    const float* __restrict__ x, const float* __restrict__ W,
    const float* __restrict__ bias, const float* __restrict__ lb,
    const float* __restrict__ rb, float* __restrict__ pot)
{
  extern __shared__ float Wt[];  // [N_][WPAD_]
  // Transpose W[k][n] -> Wt[n][k]; global reads coalesced.
  for (int idx = threadIdx.x; idx < D_ * N_; idx += 128) {
    const int k = idx >> 6;       // D index
    const int n = idx & (N_ - 1); // N index
    Wt[n * WPAD_ + k] = W[idx];
  }
  __syncthreads();

  const int  wave = threadIdx.x >> 5;
  const int  lane = threadIdx.x & 31;
  const int  half = lane >> 4;   // 0: K={0,1}; 1: K={2,3}  (ISA A/B layout)
  const int  mrow = lane & 15;
  const long m0   = (long)blockIdx.x * 64 + (long)wave * 16;

  v8f acc[4] = {v8f{}, v8f{}, v8f{}, v8f{}};
  const float* xrow = x + (m0 + mrow) * D_;

  for (int k = 0; k < D_; k += 4) {
    const int kb = k + 2 * half;
    // A fragment: 16x4 f32 (V0 = K0|K2, V1 = K1|K3 per lane-half), b64 load
    const float2 av = *(const float2*)(xrow + kb);
    v2f a; a.x = av.x; a.y = av.y;
    // B fragments: contiguous (kb, kb+1) in transposed LDS -> ds_load_b64
    v2f bf[4];
#pragma unroll
    for (int nt = 0; nt < 4; ++nt) {
      const float2 bv = *(const float2*)&Wt[(nt * 16 + mrow) * WPAD_ + kb];
      bf[nt].x = bv.x; bf[nt].y = bv.y;
    }
#pragma unroll
    for (int nt = 0; nt < 4; ++nt)
      acc[nt] = __builtin_amdgcn_wmma_f32_16x16x4_f32(
          false, a, false, bf[nt], (short)0, acc[nt], false, false);
  }

  // Epilogue: D 16x16 f32 layout -> row = m0 + r + 8*half, col = nt*16 + mrow
#pragma unroll
  for (int nt = 0; nt < 4; ++nt) {
    const int   n   = nt * 16 + mrow;
    const float bn  = bias[n];
    const float lbn = lb[n];
    const float rbn = rb[n];
#pragma unroll
    for (int r = 0; r < 8; ++r) {
      const long m = m0 + r + 8 * half;
      const int  t = (int)(m & (T_ - 1));  // tiles never cross batch boundary
      float v = acc[nt][r] + bn;
      if (t == 0)      v += lbn;
      if (t == T_ - 1) v += rbn;
      pot[m * N_ + n] = v;
    }
  }
}

// ---------------------------------------------------------------------------
// Kernel 2: Viterbi forward. One block per batch. 256 thr = 64 tags x 4 splits
// (tid = j*4 + s). trans column slice in registers; state ping-pongs in LDS;
// the 16 state reads per step are 4x ds_load_b128. Backpointers as uint8.
// ---------------------------------------------------------------------------
__global__ __launch_bounds__(256) void viterbi_fwd_kernel(
    const float* __restrict__ pot, const unsigned char* __restrict__ mask,
    const float* __restrict__ trans, unsigned char* __restrict__ bp,
    int* __restrict__ last_tag_ws, int* __restrict__ lens_ws,
    float* __restrict__ out_lens, float* __restrict__ out_dec)
{
  const int b   = blockIdx.x;
  const int tid = threadIdx.x;
  const int j   = tid >> 2;
  const int s   = tid & 3;

  __shared__ float st[2][N_];
  __shared__ int   red[256];
  __shared__ int   len_sh;

  // lens[b] = sum(mask[b,:]) -- mask is 1 byte per element (numpy bool)
  {
    int part = 0;
    const unsigned char* mb = mask + (size_t)b * T_;
    for (int k = tid; k < T_; k += 256) part += mb[k];
    red[tid] = part;
    __syncthreads();
    for (int off = 128; off > 0; off >>= 1) {
      if (tid < off) red[tid] += red[tid + off];
      __syncthreads();
    }
    if (tid == 0) {
      len_sh = red[0];
      lens_ws[b] = red[0];
      out_lens[b] = (float)red[0];
    }
    __syncthreads();
  }
  const int len = len_sh;

  // register-resident slice of trans column j: i in [s*16, s*16+16)
  float tc[16];
#pragma unroll
  for (int ii = 0; ii < 16; ++ii) tc[ii] = trans[(s * 16 + ii) * N_ + j];

  const float* potb = pot + (size_t)b * T_ * N_;
  if (s == 0) st[0][j] = potb[j];  // t = 0 (boundaries already folded in)
  __syncthreads();

  unsigned char* bpb = bp + (size_t)b * (T_ - 1) * N_;
  int cur = 0;
  for (int t = 1; t < T_; ++t) {
    if (s == 0 && t + 8 < T_)  // gfx1250 global_prefetch of upcoming pot row
      __builtin_prefetch(&potb[(size_t)(t + 8) * N_ + j], 0, 0);

    const float* so = st[cur];
    // 64 B contiguous slice -> 4x ds_load_b128
    float sv[16];
    *(float4*)&sv[0]  = *(const float4*)&so[s * 16 + 0];
    *(float4*)&sv[4]  = *(const float4*)&so[s * 16 + 4];
    *(float4*)&sv[8]  = *(const float4*)&so[s * 16 + 8];
    *(float4*)&sv[12] = *(const float4*)&so[s * 16 + 12];

    float best = -INFINITY;
    int   bi   = s * 16;
#pragma unroll
    for (int ii = 0; ii < 16; ++ii) {
      const float v = sv[ii] + tc[ii];
      if (v > best) { best = v; bi = s * 16 + ii; }
    }
    // argmax-reduce the 4 split lanes (prefer lowest index on ties = jnp.argmax)
#pragma unroll
    for (int off = 1; off < 4; off <<= 1) {
      const float ov = __shfl_xor(best, off, 4);
      const int   oi = __shfl_xor(bi, off, 4);
      if (ov > best || (ov == best && oi < bi)) { best = ov; bi = oi; }
    }
    if (s == 0) {
      bpb[(size_t)(t - 1) * N_ + j] = (unsigned char)bi;
      const float nv = potb[(size_t)t * N_ + j] + best;
      st[cur ^ 1][j] = (t < len) ? nv : so[j];
    }
    __syncthreads();
    cur ^= 1;
  }

  if (tid == 0) {
    float bv = st[cur][0];
    int   bt = 0;
    for (int jj = 1; jj < N_; ++jj) {
      const float v = st[cur][jj];
      if (v > bv) { bv = v; bt = jj; }
    }
    last_tag_ws[b] = bt;
    out_dec[(size_t)b * T_ + (T_ - 1)] = (float)bt;
  }
}

// ---------------------------------------------------------------------------
// Kernel 3: backward decode. Stage backpointers into LDS in 64 KB chunks
// (coalesced), then one lane walks the chain at LDS latency.
// ---------------------------------------------------------------------------
#define CHUNK_ 1024
__global__ __launch_bounds__(256) void viterbi_bwd_kernel(
    const unsigned char* __restrict__ bp, const int* __restrict__ last_tag_ws,
    const int* __restrict__ lens_ws, float* __restrict__ out_dec)
{
  const int b   = blockIdx.x;
  const int tid = threadIdx.x;
  __shared__ unsigned char lbp[CHUNK_ * N_];  // 64 KB

  const int len  = lens_ws[b];
  const int last = last_tag_ws[b];
  const unsigned char* bpb = bp + (size_t)b * (T_ - 1) * N_;
  float* db = out_dec + (size_t)b * T_;

  int tag = last;  // only meaningful in lane tid==0
  for (int c = (T_ - 1 + CHUNK_ - 1) / CHUNK_ - 1; c >= 0; --c) {
    const int lo = c * CHUNK_;
    int hi = (c + 1) * CHUNK_;
    if (hi > T_ - 1) hi = T_ - 1;
    const int words = (hi - lo) * (N_ / 4);
    const unsigned int* src = (const unsigned int*)(bpb + (size_t)lo * N_);
    unsigned int*       dst = (unsigned int*)lbp;
    for (int w = tid; w < words; w += 256) dst[w] = src[w];
    __syncthreads();
    if (tid == 0) {
      for (int t = hi - 1; t >= lo; --t) {
        tag = (t < len - 1) ? (int)lbp[(t - lo) * N_ + tag] : last;
        db[t] = (float)tag;
      }
    }
    __syncthreads();
  }
}

// ---------------------------------------------------------------------------
// Kernel 4: copy trans to its output slot (tuple element 4)
// ---------------------------------------------------------------------------
__global__ void copy_trans_kernel(const float* __restrict__ trans,
                                  float* __restrict__ out_trans)
{
  const int i = blockIdx.x * 256 + threadIdx.x;
  if (i < N_ * N_) out_trans[i] = trans[i];
}

// ---------------------------------------------------------------------------
extern "C" void kernel_launch(void* const* d_in, const int* in_sizes, int n_in,
                              void* d_out, int out_size, void* d_ws, size_t ws_size,
                              hipStream_t stream) {
  (void)in_sizes; (void)n_in; (void)out_size; (void)ws_size;

  const float*         x     = (const float*)d_in[0];
  const unsigned char* mask  = (const unsigned char*)d_in[1];
  const float*         W     = (const float*)d_in[2];
  const float*         bias  = (const float*)d_in[3];
  const float*         trans = (const float*)d_in[4];
  const float*         lb    = (const float*)d_in[5];
  const float*         rb    = (const float*)d_in[6];

  float* out       = (float*)d_out;
  float* out_dec   = out;                                 // B*T
  float* out_pot   = out + (size_t)B_ * T_;               // B*T*N
  float* out_lens  = out_pot + (size_t)B_ * T_ * N_;      // B
  float* out_trans = out_lens + B_;                       // N*N

  unsigned char* bp       = (unsigned char*)d_ws;         // B*(T-1)*N bytes
  int*           last_tag = (int*)((char*)d_ws + (size_t)B_ * (T_ - 1) * N_);
  int*           lens_i   = last_tag + B_;

  proj_wmma_kernel<<<(B_ * T_) / 64, 128, WT_BYTES_, stream>>>(x, W, bias, lb,
                                                               rb, out_pot);
  viterbi_fwd_kernel<<<B_, 256, 0, stream>>>(out_pot, mask, trans, bp, last_tag,
                                             lens_i, out_lens, out_dec);
  viterbi_bwd_kernel<<<B_, 256, 0, stream>>>(bp, last_tag, lens_i, out_dec);
  copy_trans_kernel<<<(N_ * N_ + 255) / 256, 256, 0, stream>>>(trans, out_trans);
}